// ROIAlign_Target_55405078119273
// MI455X (gfx1250) — compile-verified
//
#include <hip/hip_runtime.h>

// ROI-Align (tf.crop_and_resize, bilinear, extrapolation_value=0)
// features: [4,256,256,64] f32, rois: [4,64,4] f32 (x1,y1,x2,y2 in 1024-space)
// out: [4,64,56,56,64] f32
//
// Roofline: ~205 MB streaming stores + ~67 MB feature reads (fits in 192 MB
// L2) => ~12 us @ 23.3 TB/s; ~0.36 GFLOP -> pure bandwidth problem.
// CDNA5 path: per-lane global_load_async_to_lds_b128 gathers (ASYNCcnt) with
// LDS double buffering; non-temporal b128 output stores keep features in L2.
// WMMA deliberately unused: 2-nnz/row interpolation weights -> a 16x16x4 WMMA
// formulation inflates FLOPs ~20x with zero byte savings.

#define FH    256
#define FW    256
#define FC    64
#define CROPH 56
#define CROPW 56
#define NBOX  64

typedef float vf4 __attribute__((ext_vector_type(4)));  // native vector for NT store

__global__ __launch_bounds__(128, 4) void roialign_async_kernel(
    const float* __restrict__ feats,
    const float* __restrict__ rois,
    float* __restrict__ out)
{
    // [buf][pixel-group][tap 00/01/10/11][lane float4]
    __shared__ float4 smem[2][8][4][16];

    const int blk = blockIdx.x;
    const int i   = blk % CROPH;
    const int bn  = blk / CROPH;          // b*NBOX + n
    const int b   = bn / NBOX;

    const float inv_img = 1.0f / 1024.0f;
    const float x1 = rois[bn * 4 + 0] * inv_img;
    const float y1 = rois[bn * 4 + 1] * inv_img;
    const float x2 = rois[bn * 4 + 2] * inv_img;
    const float y2 = rois[bn * 4 + 3] * inv_img;

    // ys = y1*(H-1) + i*(y2-y1)*(H-1)/(crop_h-1)   (match reference eval order)
    const float ys  = y1 * 255.0f + ((float)i * (y2 - y1)) * 255.0f / 55.0f;
    const bool  yv  = (ys >= 0.0f) && (ys <= 255.0f);
    const float y0f = floorf(ys);
    const float ly  = ys - y0f;
    int y0 = (int)y0f; y0 = y0 < 0 ? 0 : (y0 > 255 ? 255 : y0);
    int yi1 = y0 + 1;  yi1 = yi1 > 255 ? 255 : yi1;

    const int tid = threadIdx.x;
    const int g   = tid >> 4;   // pixel group within block: 0..7
    const int cq  = tid & 15;   // float4 quad of the 64-channel vector

    // Low 32 bits of a generic LDS pointer == LDS byte offset (aperture rule).
    const unsigned lds_base = (unsigned)(size_t)(&smem[0][0][0][0]);

    const float* frow0 = feats + (size_t)(b * FH + y0 ) * FW * FC + cq * 4;
    const float* frow1 = feats + (size_t)(b * FH + yi1) * FW * FC + cq * 4;

    // Issue the 4 bilinear-tap gathers for column j into LDS buffer `buf`.
    auto stage = [&](int j, int buf) {
        float xs = x1 * 255.0f + ((float)j * (x2 - x1)) * 255.0f / 55.0f;
        int x0 = (int)floorf(xs); x0 = x0 < 0 ? 0 : (x0 > 255 ? 255 : x0);
        int xi1 = x0 + 1; xi1 = xi1 > 255 ? 255 : xi1;
        const float* p00 = frow0 + (size_t)x0  * FC;
        const float* p01 = frow0 + (size_t)xi1 * FC;
        const float* p10 = frow1 + (size_t)x0  * FC;
        const float* p11 = frow1 + (size_t)xi1 * FC;
        unsigned l = lds_base + (unsigned)(((buf * 8 + g) * 4) * 16 + cq) * 16u;
        // Make sure prior DS reads of the buffer being overwritten retired
        // before async writes can land (async LDS writes are unordered vs DS).
        asm volatile("s_wait_dscnt 0x0" ::: "memory");
        asm volatile("global_load_async_to_lds_b128 %0, %1, off"
                     :: "v"(l       ), "v"(p00) : "memory");
        asm volatile("global_load_async_to_lds_b128 %0, %1, off"
                     :: "v"(l + 256u), "v"(p01) : "memory");
        asm volatile("global_load_async_to_lds_b128 %0, %1, off"
                     :: "v"(l + 512u), "v"(p10) : "memory");
        asm volatile("global_load_async_to_lds_b128 %0, %1, off"
                     :: "v"(l + 768u), "v"(p11) : "memory");
    };

    stage(g, 0);  // prologue: buffer 0 holds j = g

    for (int it = 0; it < 7; ++it) {
        const int j = g + it * 8;

        if (it + 1 < 7) {
            stage(g + (it + 1) * 8, (it + 1) & 1);  // prefetch next column set
            asm volatile("s_wait_asynccnt 0x4" ::: "memory"); // current 4 done
        } else {
            asm volatile("s_wait_asynccnt 0x0" ::: "memory");
        }

        const int buf = it & 1;
        float4 v00 = smem[buf][g][0][cq];
        float4 v01 = smem[buf][g][1][cq];
        float4 v10 = smem[buf][g][2][cq];
        float4 v11 = smem[buf][g][3][cq];

        float xs = x1 * 255.0f + ((float)j * (x2 - x1)) * 255.0f / 55.0f;
        float lx = xs - floorf(xs);
        float m  = ((xs >= 0.0f) && (xs <= 255.0f) && yv) ? 1.0f : 0.0f;

        vf4 o;
        {
            float t, bt;
            t   = v00.x + (v01.x - v00.x) * lx;
            bt  = v10.x + (v11.x - v10.x) * lx;
            o.x = (t + (bt - t) * ly) * m;
            t   = v00.y + (v01.y - v00.y) * lx;
            bt  = v10.y + (v11.y - v10.y) * lx;
            o.y = (t + (bt - t) * ly) * m;
            t   = v00.z + (v01.z - v00.z) * lx;
            bt  = v10.z + (v11.z - v10.z) * lx;
            o.z = (t + (bt - t) * ly) * m;
            t   = v00.w + (v01.w - v00.w) * lx;
            bt  = v10.w + (v11.w - v10.w) * lx;
            o.w = (t + (bt - t) * ly) * m;
        }

        vf4* po = (vf4*)out +
                  ((size_t)(bn * CROPH + i) * CROPW + j) * (FC / 4) + cq;
        // Write-once output: non-temporal store (th:TH_STORE_NT) so the
        // 205 MB output stream doesn't evict the 67 MB feature set from L2.
        __builtin_nontemporal_store(o, po);
    }
}

extern "C" void kernel_launch(void* const* d_in, const int* in_sizes, int n_in,
                              void* d_out, int out_size, void* d_ws, size_t ws_size,
                              hipStream_t stream) {
    (void)in_sizes; (void)n_in; (void)out_size; (void)d_ws; (void)ws_size;
    const float* feats = (const float*)d_in[0];   // [4,256,256,64]
    const float* rois  = (const float*)d_in[1];   // [4,64,4]
    float*       out   = (float*)d_out;           // [4,64,56,56,64]

    dim3 grid(4 * NBOX * CROPH);  // 14336 blocks: one per (b, n, output row)
    roialign_async_kernel<<<grid, 128, 0, stream>>>(feats, rois, out);
}